// SplatPushModel2_68994354643554
// MI455X (gfx1250) — compile-verified
//
#include <hip/hip_runtime.h>
#include <math.h>

#define EPSF 1e-8f
#define P1 42   // buf1 pitch (48 x 42), conflict-padded
#define P2 34   // buf2 pitch (48 x 34)
#define P3 33   // buf3 pitch (32 x 33)

typedef __attribute__((ext_vector_type(2))) float v2f;
typedef __attribute__((ext_vector_type(8))) float v8f;

// Normalized 9-tap Gaussian, sigma=1.5, radius=int(3*1.5)=4 (matches jnp kernel)
__constant__ float KERN[9] = {
  0.0076144f, 0.0360750f, 0.1095861f, 0.2134445f, 0.2665598f,
  0.2134445f, 0.1095861f, 0.0360750f, 0.0076144f
};

__device__ __forceinline__ float sigmoidf_(float x) {
  return 1.0f / (1.0f + __expf(-x));
}

struct Geom { float p0x, p0y, p1x, p1y, ux, uy, L; };

__device__ __forceinline__ Geom make_geom(const float* as, const float* ae, int b) {
  Geom g;
  g.p0x = as[b * 3 + 0]; g.p0y = as[b * 3 + 1];
  g.p1x = ae[b * 3 + 0]; g.p1y = ae[b * 3 + 1];
  float dx = g.p1x - g.p0x, dy = g.p1y - g.p0y;
  g.L = sqrtf(dx * dx + dy * dy + EPSF);
  g.ux = dx / g.L; g.uy = dy / g.L;
  return g;
}

__device__ __forceinline__ float swept_at(const Geom& g, float x, float y) {
  float rx = x - g.p0x, ry = y - g.p0y;
  float t = fminf(fmaxf(rx * g.ux + ry * g.uy, 0.0f), g.L);
  float dx = rx - t * g.ux, dy = ry - t * g.uy;
  float dist = sqrtf(dx * dx + dy * dy + EPSF);
  return sigmoidf_(3.0f - dist);  // (WIDTH - dist)/SIGMA, WIDTH=3, SIGMA=1
}

// Deterministic 128-entry partial reduction (blockDim=256), result broadcast.
__device__ float reduce128(const float* __restrict__ p, float* red) {
  int t = threadIdx.x;
  red[t] = (t < 128) ? p[t] : 0.0f;
  __syncthreads();
  for (int s = 128; s > 0; s >>= 1) {
    if (t < s) red[t] += red[t + s];
    __syncthreads();
  }
  float v = red[0];
  __syncthreads();
  return v;
}

// banded Gaussian matrix element: kern[i] for 0<=i<=8 else 0 (LDS-resident kern)
__device__ __forceinline__ float bandk(const float* kernS, int i) {
  unsigned ui = (unsigned)i;
  float v = kernS[ui <= 8u ? i : 0];
  return (ui <= 8u) ? v : 0.0f;
}

// ---------------- Pass A: m_swept partials -----------------------------------
__global__ void __launch_bounds__(256) k_msw(const float* __restrict__ occ,
    const float* __restrict__ as, const float* __restrict__ ae,
    float* __restrict__ pm)
{
  __shared__ float red[256];
  int b = blockIdx.y;
  Geom g = make_geom(as, ae, b);
  const float* base = occ + ((size_t)b << 20);
  int y0 = blockIdx.x * 8;
  float xb = (float)(threadIdx.x * 4);
  float acc = 0.0f;
  for (int r = 0; r < 8; ++r) {
    int y = y0 + r;
    float fy = (float)y;
    float4 v = ((const float4*)(base + ((size_t)y << 10)))[threadIdx.x];
    acc += v.x * swept_at(g, xb + 0.0f, fy);
    acc += v.y * swept_at(g, xb + 1.0f, fy);
    acc += v.z * swept_at(g, xb + 2.0f, fy);
    acc += v.w * swept_at(g, xb + 3.0f, fy);
  }
  red[threadIdx.x] = acc;
  __syncthreads();
  for (int s = 128; s > 0; s >>= 1) {
    if (threadIdx.x < s) red[threadIdx.x] += red[threadIdx.x + s];
    __syncthreads();
  }
  if (threadIdx.x == 0) pm[b * 128 + blockIdx.x] = red[0];
}

// ---------------- Pass B: existing partials ----------------------------------
__global__ void __launch_bounds__(256) k_exist(const float* __restrict__ occ,
    const float* __restrict__ as, const float* __restrict__ ae,
    const float* __restrict__ pm, float* __restrict__ pe)
{
  __shared__ float red[256];
  int b = blockIdx.y;
  Geom g = make_geom(as, ae, b);
  float m_swept = reduce128(pm + b * 128, red);
  float pd = m_swept * (1.0f / 6.0f);
  float nx = -g.uy, ny = g.ux;
  const float* base = occ + ((size_t)b << 20);
  int y0 = blockIdx.x * 8;
  float xb = (float)(threadIdx.x * 4);
  float acc = 0.0f;
  for (int r = 0; r < 8; ++r) {
    int y = y0 + r;
    float fy = (float)y;
    float4 v = ((const float4*)(base + ((size_t)y << 10)))[threadIdx.x];
    float qy = fy - g.p1y;
    #pragma unroll
    for (int i = 0; i < 4; ++i) {
      float qx = (xb + (float)i) - g.p1x;
      float s = qx * g.ux + qy * g.uy;
      float rr = qx * nx + qy * ny;
      float lat = sigmoidf_(3.0f - fabsf(rr));
      float probe = sigmoidf_(s) * sigmoidf_(pd - s) * lat;
      acc += (&v.x)[i] * probe;
    }
  }
  red[threadIdx.x] = acc;
  __syncthreads();
  for (int s = 128; s > 0; s >>= 1) {
    if (threadIdx.x < s) red[threadIdx.x] += red[threadIdx.x + s];
    __syncthreads();
  }
  if (threadIdx.x == 0) pe[b * 128 + blockIdx.x] = red[0];
}

// ---------------- Pass C: sum of dep_mask (no data reads) --------------------
__global__ void __launch_bounds__(256) k_dm(
    const float* __restrict__ as, const float* __restrict__ ae,
    const float* __restrict__ pm, const float* __restrict__ pe,
    float* __restrict__ pdm)
{
  __shared__ float red[256];
  int b = blockIdx.y;
  Geom g = make_geom(as, ae, b);
  float m_swept = reduce128(pm + b * 128, red);
  float existing = reduce128(pe + b * 128, red);
  float pd = m_swept * (1.0f / 6.0f);
  float extra = existing * (1.0f / 6.0f);
  float nx = -g.uy, ny = g.ux;
  int y0 = blockIdx.x * 8;
  float xb = (float)(threadIdx.x * 4);
  float acc = 0.0f;
  for (int r = 0; r < 8; ++r) {
    float fy = (float)(y0 + r);
    float qy = fy - g.p1y;
    #pragma unroll
    for (int i = 0; i < 4; ++i) {
      float qx = (xb + (float)i) - g.p1x;
      float s = qx * g.ux + qy * g.uy;
      float rr = qx * nx + qy * ny;
      float lat = sigmoidf_(3.0f - fabsf(rr));
      acc += sigmoidf_(s - extra) * sigmoidf_(extra + pd - s) * lat;
    }
  }
  red[threadIdx.x] = acc;
  __syncthreads();
  for (int s = 128; s > 0; s >>= 1) {
    if (threadIdx.x < s) red[threadIdx.x] += red[threadIdx.x + s];
    __syncthreads();
  }
  if (threadIdx.x == 0) pdm[b * 128 + blockIdx.x] = red[0];
}

// ---------------- Pass D: dep field + WMMA separable blur + combine ----------
// 32x32 output tile; extended 40x40 analytic dep field; blur as banded matmul
// using V_WMMA_F32_16X16X4_F32 (D = A(16x24) x W(24x16) in K=4 chunks).
__global__ void __launch_bounds__(256) k_final(const float* __restrict__ occ,
    const float* __restrict__ as, const float* __restrict__ ae,
    const float* __restrict__ pm, const float* __restrict__ pe,
    const float* __restrict__ pdm, float* __restrict__ out)
{
  __shared__ float red[256];
  __shared__ float kernS[9];
  __shared__ float buf1[48 * P1];  // dep: rows = ext X (0..39 valid, 40..47 zero), cols = ext Y (40)
  __shared__ float buf2[48 * P2];  // Y-blurred: rows = ext X, cols = out Y (32)
  __shared__ float buf3[32 * P3];  // fully blurred: rows = out X, cols = out Y

  int t = threadIdx.x;
  int b = blockIdx.z;
  int x0 = blockIdx.x * 32, y0 = blockIdx.y * 32;

  Geom g = make_geom(as, ae, b);
  float nx = -g.uy, ny = g.ux;
  float m_swept  = reduce128(pm  + b * 128, red);
  float existing = reduce128(pe  + b * 128, red);
  float sum_dm   = reduce128(pdm + b * 128, red);
  float pd = m_swept * (1.0f / 6.0f);
  float extra = existing * (1.0f / 6.0f);
  float dep_norm = m_swept / (sum_dm + EPSF);
  if (t < 9) kernS[t] = KERN[t];

  // Stage 1: analytic dep on extended tile (zero outside domain -> SAME conv)
  for (int p = t; p < 48 * 40; p += 256) {
    int i = p / 40, j = p - i * 40;     // i: ext X row, j: ext Y col
    float val = 0.0f;
    int gx = x0 - 4 + i, gy = y0 - 4 + j;
    if (i < 40 && gx >= 0 && gx < 1024 && gy >= 0 && gy < 1024) {
      float qx = (float)gx - g.p1x, qy = (float)gy - g.p1y;
      float s = qx * g.ux + qy * g.uy;
      float rr = qx * nx + qy * ny;
      float lat = sigmoidf_(3.0f - fabsf(rr));
      val = sigmoidf_(s - extra) * sigmoidf_(extra + pd - s) * lat * dep_norm;
    }
    buf1[i * P1 + j] = val;
  }
  __syncthreads();

  int wave = t >> 5, lane = t & 31;
  int m = lane & 15, kh = lane >> 4;   // ISA 7.12.2 fragment indices (wave32)

  // Stage 2: Y-blur. D(16x,16y) = A(dep 16x x 24yext) * Wy(24x16 banded)
  if (wave < 6) {                      // 3 row-chunks (48 rows) x 2 col-chunks
    int rc = wave >> 1, cc = wave & 1;
    const float* Arow = &buf1[(rc * 16 + m) * P1 + cc * 16];
    v8f acc = {0.f, 0.f, 0.f, 0.f, 0.f, 0.f, 0.f, 0.f};
    #pragma unroll
    for (int ch = 0; ch < 6; ++ch) {
      int k0 = ch * 4 + 2 * kh;        // A: VGPR0 holds K={0,2}+4ch, VGPR1 K={1,3}+4ch
      v2f a;  a.x = Arow[k0];              a.y = Arow[k0 + 1];
      v2f bb; bb.x = bandk(kernS, k0 - m); bb.y = bandk(kernS, k0 + 1 - m);
      acc = __builtin_amdgcn_wmma_f32_16x16x4_f32(false, a, false, bb,
                                                  (short)0, acc, false, false);
    }
    #pragma unroll
    for (int v = 0; v < 8; ++v)        // D: VGPR v -> row v (+8 for upper half-wave)
      buf2[(rc * 16 + v + 8 * kh) * P2 + cc * 16 + m] = acc[v];
  }
  __syncthreads();

  // Stage 3: X-blur. D(16xout,16y) = Wx(16x24 banded) * T(24xext x 16y)
  if (wave < 4) {                      // 2 x 2 output 16x16 tiles
    int mc = wave >> 1, nc = wave & 1;
    v8f acc = {0.f, 0.f, 0.f, 0.f, 0.f, 0.f, 0.f, 0.f};
    #pragma unroll
    for (int ch = 0; ch < 6; ++ch) {
      int k0 = ch * 4 + 2 * kh;
      v2f a;  a.x = bandk(kernS, k0 - m);  a.y = bandk(kernS, k0 + 1 - m);
      v2f bb; bb.x = buf2[(mc * 16 + k0)     * P2 + nc * 16 + m];
              bb.y = buf2[(mc * 16 + k0 + 1) * P2 + nc * 16 + m];
      acc = __builtin_amdgcn_wmma_f32_16x16x4_f32(false, a, false, bb,
                                                  (short)0, acc, false, false);
    }
    #pragma unroll
    for (int v = 0; v < 8; ++v)
      buf3[(mc * 16 + v + 8 * kh) * P3 + nc * 16 + m] = acc[v];
  }
  __syncthreads();

  // Stage 4: out[b,y,x] = occ[b,y,x]*(1-swept(x,y)) + blurred_dep(x,y)
  {
    int yl = t >> 3;           // 0..31
    int xl = (t & 7) * 4;      // 0,4,...,28 (float4 along contiguous x)
    int gy = y0 + yl;
    size_t off = ((size_t)b << 20) + ((size_t)gy << 10) + (size_t)(x0 + xl);
    float4 vin = *((const float4*)(occ + off));
    float4 vout;
    #pragma unroll
    for (int i = 0; i < 4; ++i) {
      float x = (float)(x0 + xl + i);
      float sw = swept_at(g, x, (float)gy);
      (&vout.x)[i] = (&vin.x)[i] * (1.0f - sw) + buf3[(xl + i) * P3 + yl];
    }
    *((float4*)(out + off)) = vout;
  }
}

extern "C" void kernel_launch(void* const* d_in, const int* in_sizes, int n_in,
                              void* d_out, int out_size, void* d_ws, size_t ws_size,
                              hipStream_t stream) {
  (void)n_in; (void)out_size; (void)ws_size;
  const float* occ = (const float*)d_in[0];
  const float* as  = (const float*)d_in[1];
  const float* ae  = (const float*)d_in[2];
  float* out = (float*)d_out;
  float* ws  = (float*)d_ws;
  int B = in_sizes[1] / 3;           // 16

  float* pm  = ws;                   // [B][128] m_swept partials
  float* pe  = ws + 2048;            // [B][128] existing partials
  float* pdm = ws + 4096;            // [B][128] sum(dep_mask) partials

  dim3 rgrid(128, B);
  k_msw  <<<rgrid, 256, 0, stream>>>(occ, as, ae, pm);
  k_exist<<<rgrid, 256, 0, stream>>>(occ, as, ae, pm, pe);
  k_dm   <<<rgrid, 256, 0, stream>>>(as, ae, pm, pe, pdm);
  dim3 fgrid(32, 32, B);
  k_final<<<fgrid, 256, 0, stream>>>(occ, as, ae, pm, pe, pdm, out);
}